// SpikingLayer_81484119540334
// MI455X (gfx1250) — compile-verified
//
#include <hip/hip_runtime.h>

typedef float v2f __attribute__((ext_vector_type(2)));
typedef float v8f __attribute__((ext_vector_type(8)));

#define TC      128     // time steps per chunk
#define NCH     16      // channels per block (one WMMA N tile)
#define WIN     228     // spike window per chunk: TC + 100
#define LK      100     // kernel length
#define THREADS 256     // 8 wave32 -> 8 x 16 time rows = TC

__global__ __launch_bounds__(THREADS)
void slayer_fused(const float* __restrict__ spikes,  // (1024,16,2,2048)
                  const float* __restrict__ epsp,    // (2,100)
                  const float* __restrict__ refk,    // (100) unused (analytic)
                  float* __restrict__ out)           // (1024,16,2048)
{
  __shared__ float xwin[2][WIN][NCH];   // spike window, [syn][t'][ch]
  __shared__ float vbuf[TC][NCH];       // vmem chunk for the scan
  __shared__ float ring[128][NCH];      // past spike counts (truncation fix)
  __shared__ float krev[2][160];        // REVERSED zero-padded EPSP kernel:
                                        // krev[s][j] = epsp[s][127-j] for j in [28,128)

  const int tid  = threadIdx.x;
  const int lane = tid & 31;
  const int wid  = tid >> 5;          // wave id 0..7 = time tile
  const int lm   = lane & 15;
  const int lg   = lane >> 4;         // lane group (0/1)

  const int c0 = blockIdx.x * NCH;    // channel base, c = b*2048 + n
  const int b  = c0 >> 11;
  const int n0 = c0 & 2047;

  // reversed kernel image so A-fragment loads are ascending-consecutive
  for (int i = tid; i < 2 * 160; i += THREADS) {
    int s = i / 160, j = i % 160;
    float v = 0.f;
    if (j >= 28 && j < 128) v = epsp[s * LK + (127 - j)];
    krev[s][j] = v;
  }
  for (int i = tid; i < 128 * NCH; i += THREADS)
    (&ring[0][0])[i] = 0.f;

  // refractory recursion constants: d = exp(-dt/tau_mem), d^100
  const float dDec = 0.9048374180359595f;
  const float d100 = 4.5399929762484854e-05f;
  float R = 0.f;                      // live in scan lanes only

  const int SN  = 2 * 2048;           // stride per (t,b)
  const int BSN = 16 * SN;            // stride per t
  (void)refk;

  for (int cc = 0; cc < 1024 / TC; ++cc) {
    const int cb = cc * TC;

    // ---- load spike window t' in [cb-100, cb+127] (zeros for t'<0) ----
    for (int i = tid; i < WIN * 2 * 4; i += THREADS) {
      int q  = i & 3;                 // float4 within 16-ch row
      int sr = i >> 2;
      int s  = sr & 1;
      int w  = sr >> 1;               // 0..227
      int tt = cb - 100 + w;
      float4 v = make_float4(0.f, 0.f, 0.f, 0.f);
      if (tt >= 0) {
        const float* p = spikes + (size_t)tt * BSN + b * SN + s * 2048 + n0 + 4 * q;
        v = *(const float4*)p;
      }
      *(float4*)&xwin[s][w][4 * q] = v;
    }
    __syncthreads();

    // ---- banded-Toeplitz conv via V_WMMA_F32_16X16X4_F32 ----
    // D[m][n] = sum_{t'} ker[(t0+m)-t'] * x[t'][n];  K steps of 4 over the band
    // A[m][K] = ker[m + 100 - 4*it - K]; reversed image: index j0 = 27-lm+2lg+4it
    v8f acc = {};
    const int t0off = 16 * wid;               // tile's time offset in chunk
    const int jb = 27 - lm + 2 * lg;          // reversed-kernel index base
    for (int it = 0; it < 29; ++it) {
      const int w0 = t0off + 4 * it + 2 * lg; // window row for this lane's K0
      const int j0 = jb + 4 * it;
      #pragma unroll
      for (int s = 0; s < 2; ++s) {
        v2f a, bv;
        a.x  = krev[s][j0];                   // A: VGPR0 -> K0   (tap kidx)
        a.y  = krev[s][j0 + 1];               // A: VGPR1 -> K0+1 (tap kidx-1)
        bv.x = xwin[s][w0][lm];               // B: VGPR0 -> K0
        bv.y = xwin[s][w0 + 1][lm];           // B: VGPR1 -> K0+1
        acc = __builtin_amdgcn_wmma_f32_16x16x4_f32(
            false, a, false, bv, (short)0, acc, false, false);
      }
    }
    #pragma unroll
    for (int j = 0; j < 8; ++j)               // C/D layout: M = j + 8*lg, N = lm
      vbuf[t0off + j + 8 * lg][lm] = acc[j];
    __syncthreads();

    // ---- sequential threshold + refractory scan (16 lanes) ----
    if (tid < NCH) {
      const int ch = tid;
      float r = R;
      for (int tl = 0; tl < TC; ++tl) {
        const int t = cb + tl;
        float veff = vbuf[tl][ch] + r;
        float n    = floorf(fmaxf(veff, 0.f));           // threshold = 1
        float n99  = ring[(t + 29) & 127][ch];           // n(t-99)
        r = dDec * (r - n) + d100 * n99;                 // exact-truncation R
        ring[t & 127][ch] = n;
        // straight-through value: (n - sg*veff) + sg*veff
        float sg   = 2.f * __expf(-2.f * fabsf(veff - 1.f));
        float term = sg * veff;
        out[(size_t)t * (16 * 2048) + c0 + ch] = (n - term) + term;
      }
      R = r;
    }
    // next chunk's post-load barrier orders scan reads vs. vbuf rewrites
  }
}

extern "C" void kernel_launch(void* const* d_in, const int* in_sizes, int n_in,
                              void* d_out, int out_size, void* d_ws, size_t ws_size,
                              hipStream_t stream) {
  const float* spikes = (const float*)d_in[0];  // (1024,16,2,2048) f32
  const float* epsp   = (const float*)d_in[1];  // (2,100) f32
  const float* refk   = (const float*)d_in[2];  // (100) f32
  float* out = (float*)d_out;                   // (1024,16,2048) f32
  (void)in_sizes; (void)n_in; (void)out_size; (void)d_ws; (void)ws_size;

  const int BN = 16 * 2048;
  dim3 grid(BN / NCH);      // 2048 blocks, each owns 16 channels for all T
  dim3 block(THREADS);
  hipLaunchKernelGGL(slayer_fused, grid, block, 0, stream,
                     spikes, epsp, refk, out);
}